// GraphSAGE_87892210745360
// MI455X (gfx1250) — compile-verified
//
#include <hip/hip_runtime.h>
#include <hip/hip_bf16.h>

typedef __attribute__((ext_vector_type(2))) float v2f;
typedef __attribute__((ext_vector_type(8))) float v8f;

#define FD 128      // feature dim (D == H == 128)
#define LDA2 132    // LDS row stride in floats: 16B-aligned rows, banks (4m+k)%64 conflict-free
#define NCLS 47

// ---------------- zero fill ----------------
__global__ void zero_f32(float* __restrict__ p, size_t n) {
    size_t i = (size_t)blockIdx.x * blockDim.x + threadIdx.x;
    size_t stride = (size_t)gridDim.x * blockDim.x;
    for (; i < n; i += stride) p[i] = 0.0f;
}

// ---------------- weight transpose: WT[c*R + r] = W[r*C + c] ----------------
__global__ void transpose_kernel(const float* __restrict__ W, float* __restrict__ WT,
                                 int R, int C) {
    int i = blockIdx.x * blockDim.x + threadIdx.x;
    int n = R * C;
    int stride = gridDim.x * blockDim.x;
    for (; i < n; i += stride) {
        int r = i / C, c = i - r * C;
        WT[(size_t)c * R + r] = W[i];
    }
}

// ---------------- degree ----------------
__global__ void degree_kernel(const int* __restrict__ dst, float* __restrict__ deg, int E) {
    int i = blockIdx.x * blockDim.x + threadIdx.x;
    int stride = gridDim.x * blockDim.x;
    for (; i < E; i += stride) atomicAdd(&deg[dst[i]], 1.0f);
}

// ---------------- edge gather + scatter-add (wave per edge) ----------------
__global__ void scatter_kernel(const float* __restrict__ X, const int* __restrict__ src,
                               const int* __restrict__ dst, float* __restrict__ agg, int E) {
    int gwaves = (gridDim.x * blockDim.x) >> 5;
    int wid = (blockIdx.x * blockDim.x + threadIdx.x) >> 5;
    int lane = threadIdx.x & 31;
    for (int e = wid; e < E; e += gwaves) {
        int s = src[e];
        int d = dst[e];
        int en = e + gwaves;
        if (en < E) __builtin_prefetch(X + (size_t)src[en] * FD + lane * 4, 0, 0);
        const float4* row = (const float4*)(X + (size_t)s * FD);
        float4 v = row[lane];
        float* out = agg + (size_t)d * FD + lane * 4;
        atomicAdd(out + 0, v.x);
        atomicAdd(out + 1, v.y);
        atomicAdd(out + 2, v.z);
        atomicAdd(out + 3, v.w);
    }
}

// LDS byte offset of a __shared__ object (generic shared ptr low 32 bits == LDS addr)
__device__ __forceinline__ unsigned lds_off(const void* p) {
    return (unsigned)(unsigned long long)p;
}

__device__ __forceinline__ void async_b128_to_lds(unsigned lds_addr, const void* gaddr) {
    asm volatile("global_load_async_to_lds_b128 %0, %1, off"
                 :: "v"(lds_addr), "v"(gaddr) : "memory");
}

// ---------------- SAGE layer: relu(X@Wself + (agg/deg)@Wneigh + b) ----------------
__global__ void __launch_bounds__(256)
sage_layer_kernel(const float* __restrict__ Xin, const float* __restrict__ agg,
                  const float* __restrict__ deg, const float* __restrict__ WselfT,
                  const float* __restrict__ WneighT, const float* __restrict__ bias,
                  float* __restrict__ Out, int N) {
    __shared__ float sA[16 * LDA2];   // self-feature tile (raw)
    __shared__ float sNB[16 * LDA2];  // neighbor-sum tile (raw; scaled at read)
    __shared__ float sInv[16];        // 1 / max(deg,1)

    int rowBase = blockIdx.x * 16;
    int t = threadIdx.x;

    if (t < 16) {
        int r = rowBase + t; if (r > N - 1) r = N - 1;
        sInv[t] = 1.0f / fmaxf(deg[r], 1.0f);
    }

    // 16x128 floats = 512 float4 per tile; 256 threads -> 2 float4 each per tile
    for (int i = t; i < 512; i += 256) {
        int r = i >> 5;         // 32 float4 per row
        int c4 = i & 31;
        int gr = rowBase + r; if (gr > N - 1) gr = N - 1;
        async_b128_to_lds(lds_off(&sA[r * LDA2 + c4 * 4]),  Xin + (size_t)gr * FD + c4 * 4);
        async_b128_to_lds(lds_off(&sNB[r * LDA2 + c4 * 4]), agg + (size_t)gr * FD + c4 * 4);
    }
    asm volatile("s_wait_asynccnt 0" ::: "memory");
    __syncthreads();

    int wave = t >> 5, lane = t & 31;
    int ln15 = lane & 15;            // A: row-in-tile m; B/C: col-in-tile n
    int kk = (lane >> 4) << 1;       // lanes 0-15 -> K pair {0,1}; lanes 16-31 -> {2,3}
    int colBase = wave << 4;
    float inv = sInv[ln15];          // neighbor-mean scale for A rows

    const v2f* bSelf  = (const v2f*)(WselfT  + (size_t)(colBase + ln15) * FD);
    const v2f* bNeigh = (const v2f*)(WneighT + (size_t)(colBase + ln15) * FD);

    v8f acc = {};
    for (int k0 = 0; k0 < FD; k0 += 4) {
        int kidx = (k0 + kk) >> 1;
        v2f a, b;
        a[0] = sA[ln15 * LDA2 + k0 + kk];
        a[1] = sA[ln15 * LDA2 + k0 + kk + 1];
        b = bSelf[kidx];
        acc = __builtin_amdgcn_wmma_f32_16x16x4_f32(false, a, false, b,
                                                    (short)0, acc, false, false);
        a[0] = sNB[ln15 * LDA2 + k0 + kk] * inv;
        a[1] = sNB[ln15 * LDA2 + k0 + kk + 1] * inv;
        b = bNeigh[kidx];
        acc = __builtin_amdgcn_wmma_f32_16x16x4_f32(false, a, false, b,
                                                    (short)0, acc, false, false);
    }

    float bn = bias[colBase + ln15];
    int mHi = (lane >> 4) << 3;
#pragma unroll
    for (int v = 0; v < 8; ++v) {
        int gr = rowBase + v + mHi;
        if (gr < N) {
            float val = fmaxf(acc[v] + bn, 0.0f);
            Out[(size_t)gr * FD + colBase + ln15] = val;
        }
    }
}

// ---------------- FC head: H@Wfc + bfc (128 -> 47, padded to 48 cols) ----------------
__global__ void __launch_bounds__(96)
fc_kernel(const float* __restrict__ Hin, const float* __restrict__ WfcT,
          const float* __restrict__ bfc, float* __restrict__ Out, int N) {
    __shared__ float sA[16 * LDA2];
    int rowBase = blockIdx.x * 16;
    int t = threadIdx.x;

    for (int i = t; i < 512; i += 96) {
        int r = i >> 5, c4 = i & 31;
        int gr = rowBase + r; if (gr > N - 1) gr = N - 1;
        async_b128_to_lds(lds_off(&sA[r * LDA2 + c4 * 4]), Hin + (size_t)gr * FD + c4 * 4);
    }
    asm volatile("s_wait_asynccnt 0" ::: "memory");
    __syncthreads();

    int wave = t >> 5, lane = t & 31;
    int ln15 = lane & 15;
    int kk = (lane >> 4) << 1;
    int col = wave * 16 + ln15;
    const v2f* bFc = (const v2f*)(WfcT + (size_t)(col < NCLS ? col : 0) * FD);
    bool live = (col < NCLS);

    v8f acc = {};
    for (int k0 = 0; k0 < FD; k0 += 4) {
        int kidx = (k0 + kk) >> 1;
        v2f a, b;
        a[0] = sA[ln15 * LDA2 + k0 + kk];
        a[1] = sA[ln15 * LDA2 + k0 + kk + 1];
        b = bFc[kidx];
        if (!live) { b[0] = 0.0f; b[1] = 0.0f; }
        acc = __builtin_amdgcn_wmma_f32_16x16x4_f32(false, a, false, b,
                                                    (short)0, acc, false, false);
    }

    if (live) {
        float bn = bfc[col];
        int mHi = (lane >> 4) << 3;
#pragma unroll
        for (int v = 0; v < 8; ++v) {
            int gr = rowBase + v + mHi;
            if (gr < N) Out[(size_t)gr * NCLS + col] = acc[v] + bn;
        }
    }
}

// ---------------- launch ----------------
extern "C" void kernel_launch(void* const* d_in, const int* in_sizes, int n_in,
                              void* d_out, int out_size, void* d_ws, size_t ws_size,
                              hipStream_t stream) {
    const float* feat   = (const float*)d_in[0];
    const int*   src    = (const int*)d_in[1];
    const int*   dst    = (const int*)d_in[2];
    const float* Wself0 = (const float*)d_in[3];
    const float* Wneigh0= (const float*)d_in[4];
    const float* b0     = (const float*)d_in[5];
    const float* Wself1 = (const float*)d_in[6];
    const float* Wneigh1= (const float*)d_in[7];
    const float* b1     = (const float*)d_in[8];
    const float* Wfc    = (const float*)d_in[9];
    const float* bfc    = (const float*)d_in[10];
    float* out = (float*)d_out;

    int N = in_sizes[0] / FD;
    int E = in_sizes[1];

    // ws: deg[N] | agg[N*128] | h0[N*128] | h1[N*128] | 4x WT(128x128) | WTfc(47x128)
    float* ws   = (float*)d_ws;
    float* deg  = ws;
    float* agg  = ws + N;
    float* h0   = agg + (size_t)N * FD;
    float* h1   = h0 + (size_t)N * FD;
    float* WT0s = h1 + (size_t)N * FD;
    float* WT0n = WT0s + FD * FD;
    float* WT1s = WT0n + FD * FD;
    float* WT1n = WT1s + FD * FD;
    float* WTfc = WT1n + FD * FD;

    int rowTiles = (N + 15) / 16;

    zero_f32<<<2048, 256, 0, stream>>>(deg, (size_t)N * (FD + 1));
    transpose_kernel<<<64, 256, 0, stream>>>(Wself0,  WT0s, FD, FD);
    transpose_kernel<<<64, 256, 0, stream>>>(Wneigh0, WT0n, FD, FD);
    transpose_kernel<<<64, 256, 0, stream>>>(Wself1,  WT1s, FD, FD);
    transpose_kernel<<<64, 256, 0, stream>>>(Wneigh1, WT1n, FD, FD);
    transpose_kernel<<<24, 256, 0, stream>>>(Wfc,     WTfc, FD, NCLS);
    degree_kernel<<<1024, 256, 0, stream>>>(dst, deg, E);

    // layer 0
    scatter_kernel<<<4096, 256, 0, stream>>>(feat, src, dst, agg, E);
    sage_layer_kernel<<<rowTiles, 256, 0, stream>>>(feat, agg, deg, WT0s, WT0n, b0, h0, N);

    // layer 1
    zero_f32<<<2048, 256, 0, stream>>>(agg, (size_t)N * FD);
    scatter_kernel<<<4096, 256, 0, stream>>>(h0, src, dst, agg, E);
    sage_layer_kernel<<<rowTiles, 256, 0, stream>>>(h0, agg, deg, WT1s, WT1n, b1, h1, N);

    // FC head
    fc_kernel<<<rowTiles, 96, 0, stream>>>(h1, WTfc, bfc, out, N);
}